// GCN_4020089389121
// MI455X (gfx1250) — compile-verified
//
#include <hip/hip_runtime.h>
#include <math.h>

// FP32 WMMA vector types (CDNA5 / gfx1250, wave32)
typedef float v2f __attribute__((ext_vector_type(2)));
typedef float v8f __attribute__((ext_vector_type(8)));

#define NFEAT 128   // feature dim of embeddings / h
#define NOUT  64    // output logits dim

// ---------------------------------------------------------------------------
// zero a float buffer
// ---------------------------------------------------------------------------
__global__ void gcn_zero_kernel(float* __restrict__ p, int n) {
    int i = blockIdx.x * blockDim.x + threadIdx.x;
    if (i < n) p[i] = 0.0f;
}

// ---------------------------------------------------------------------------
// SpMM: out[rows[e], :] += vals[e] * x[cols[e], :]   (d = 128)
// One wave32 per edge: each lane moves a float4 (32 lanes * 16B = 512B row).
// Gather is coalesced; scatter uses global_atomic_add_f32 (L2-resident).
// ---------------------------------------------------------------------------
__global__ void gcn_spmm_kernel(const int* __restrict__ rows,
                                const int* __restrict__ cols,
                                const float* __restrict__ vals,
                                const float* __restrict__ x,
                                float* __restrict__ out,
                                int n_edges) {
    int wave = (blockIdx.x * (blockDim.x >> 5)) + (threadIdx.x >> 5);
    int lane = threadIdx.x & 31;
    if (wave >= n_edges) return;

    int r = rows[wave];          // scalar (uniform) loads within the wave
    int c = cols[wave];
    float v = vals[wave];

    const float4* xr = (const float4*)(x + (size_t)c * NFEAT);
    float*        o  = out + (size_t)r * NFEAT;

    float4 d = xr[lane];
    int base = lane * 4;
    atomicAdd(&o[base + 0], v * d.x);
    atomicAdd(&o[base + 1], v * d.y);
    atomicAdd(&o[base + 2], v * d.z);
    atomicAdd(&o[base + 3], v * d.w);
}

// ---------------------------------------------------------------------------
// GEMM1 + ReLU: Out[50000,128] = relu(A[50000,128] @ W[128,128])
// fp32 WMMA 16x16x4. Block = 256 threads = 8 waves; block b owns 16-row
// panel m0 = 16*b, wave w owns N-tile n0 = 16*w. K loop: 32 WMMAs.
// ---------------------------------------------------------------------------
__global__ void gcn_gemm1_relu_kernel(const float* __restrict__ A,
                                      const float* __restrict__ W,
                                      float* __restrict__ Out) {
    int m0   = blockIdx.x * 16;
    int wv   = threadIdx.x >> 5;       // 0..7 -> N tile
    int lane = threadIdx.x & 31;
    int n0   = wv * 16;

    int mrow  = lane & 15;             // M (for A) / N (for B) within tile
    int khalf = (lane >> 4) * 2;       // lanes 16..31 hold K=2,3

    const float* arow = A + (size_t)(m0 + mrow) * NFEAT;

    v8f acc = {};
    #pragma unroll 4
    for (int k = 0; k < NFEAT; k += 4) {
        v2f a, b;
        a.x = arow[k + khalf + 0];
        a.y = arow[k + khalf + 1];
        b.x = W[(size_t)(k + khalf + 0) * NFEAT + n0 + mrow];
        b.y = W[(size_t)(k + khalf + 1) * NFEAT + n0 + mrow];
        acc = __builtin_amdgcn_wmma_f32_16x16x4_f32(
            /*neg_a=*/false, a, /*neg_b=*/false, b,
            /*c_mod=*/(short)0, acc, /*reuse_a=*/false, /*reuse_b=*/false);
    }

    // C/D layout: VGPR r -> M = r + 8*(lane>=16), N = lane&15
    int ncol  = n0 + (lane & 15);
    int mbase = m0 + ((lane >> 4) * 8);
    #pragma unroll
    for (int r = 0; r < 8; ++r) {
        float v = acc[r];
        Out[(size_t)(mbase + r) * NFEAT + ncol] = v > 0.0f ? v : 0.0f;
    }
}

// ---------------------------------------------------------------------------
// GEMM2 + softmax: Out[50000,64] = softmax(A[50000,128] @ W[128,64], axis=1)
// Block = 128 threads = 4 waves; each wave computes one 16x16 tile of the
// 16x64 logits panel via fp32 WMMA, stages to LDS, then 8 threads per row
// do a numerically-stable softmax with width-8 shuffles.
// ---------------------------------------------------------------------------
__global__ void gcn_gemm2_softmax_kernel(const float* __restrict__ A,
                                         const float* __restrict__ W,
                                         float* __restrict__ Out) {
    __shared__ float lds[16 * NOUT];

    int m0   = blockIdx.x * 16;
    int wv   = threadIdx.x >> 5;       // 0..3 -> N tile
    int lane = threadIdx.x & 31;
    int n0   = wv * 16;

    int mrow  = lane & 15;
    int khalf = (lane >> 4) * 2;

    const float* arow = A + (size_t)(m0 + mrow) * NFEAT;

    v8f acc = {};
    #pragma unroll 4
    for (int k = 0; k < NFEAT; k += 4) {
        v2f a, b;
        a.x = arow[k + khalf + 0];
        a.y = arow[k + khalf + 1];
        b.x = W[(size_t)(k + khalf + 0) * NOUT + n0 + mrow];
        b.y = W[(size_t)(k + khalf + 1) * NOUT + n0 + mrow];
        acc = __builtin_amdgcn_wmma_f32_16x16x4_f32(
            false, a, false, b, (short)0, acc, false, false);
    }

    int ncol  = n0 + (lane & 15);
    int rbase = (lane >> 4) * 8;
    #pragma unroll
    for (int r = 0; r < 8; ++r)
        lds[(rbase + r) * NOUT + ncol] = acc[r];

    __syncthreads();

    // softmax: 128 threads -> 16 rows x 8 threads/row (groups stay in-wave)
    int row = threadIdx.x >> 3;        // 0..15
    int t   = threadIdx.x & 7;         // 0..7
    const float* lrow = &lds[row * NOUT];

    float mx = -3.402823466e+38f;
    #pragma unroll
    for (int j = t; j < NOUT; j += 8) mx = fmaxf(mx, lrow[j]);
    #pragma unroll
    for (int off = 4; off >= 1; off >>= 1) mx = fmaxf(mx, __shfl_xor(mx, off, 8));

    float s = 0.0f;
    #pragma unroll
    for (int j = t; j < NOUT; j += 8) s += __expf(lrow[j] - mx);
    #pragma unroll
    for (int off = 4; off >= 1; off >>= 1) s += __shfl_xor(s, off, 8);

    float inv = 1.0f / s;
    float* orow = Out + (size_t)(m0 + row) * NOUT;
    #pragma unroll
    for (int j = t; j < NOUT; j += 8) orow[j] = __expf(lrow[j] - mx) * inv;
}

// ---------------------------------------------------------------------------
// launch
// ---------------------------------------------------------------------------
extern "C" void kernel_launch(void* const* d_in, const int* in_sizes, int n_in,
                              void* d_out, int out_size, void* d_ws, size_t ws_size,
                              hipStream_t stream) {
    const float* emb  = (const float*)d_in[0];   // [N,128]
    const int*   rows = (const int*)  d_in[1];   // [E]
    const int*   cols = (const int*)  d_in[2];   // [E]
    const float* vals = (const float*)d_in[3];   // [E]
    const float* W1   = (const float*)d_in[4];   // [128,128]
    const float* W2   = (const float*)d_in[5];   // [128,64]
    float*       out  = (float*)d_out;           // [N,64]

    const int E = in_sizes[1];
    const int N = in_sizes[0] / NFEAT;           // 50000 (== 3125 * 16)
    const int NE = N * NFEAT;

    float* buf0 = (float*)d_ws;                  // agg buffer (reused layer 2)
    float* buf1 = buf0 + (size_t)NE;             // h = relu(agg @ W1)

    const int mtiles      = N / 16;              // 3125
    const int zero_blocks = (NE + 255) / 256;
    const int spmm_blocks = (E + 7) / 8;         // 8 waves (edges) per 256-thr block

    // layer 1
    gcn_zero_kernel<<<zero_blocks, 256, 0, stream>>>(buf0, NE);
    gcn_spmm_kernel<<<spmm_blocks, 256, 0, stream>>>(rows, cols, vals, emb, buf0, E);
    gcn_gemm1_relu_kernel<<<mtiles, 256, 0, stream>>>(buf0, W1, buf1);

    // layer 2 (reuse buf0 for the second aggregation)
    gcn_zero_kernel<<<zero_blocks, 256, 0, stream>>>(buf0, NE);
    gcn_spmm_kernel<<<spmm_blocks, 256, 0, stream>>>(rows, cols, vals, buf1, buf0, E);
    gcn_gemm2_softmax_kernel<<<mtiles, 128, 0, stream>>>(buf0, W2, out);
}